// _NonLocalBlockND_89429809037749
// MI455X (gfx1250) — compile-verified
//
#include <hip/hip_runtime.h>
#include <hip/hip_bf16.h>
#include <math.h>

typedef __attribute__((ext_vector_type(2))) float v2f;
typedef __attribute__((ext_vector_type(8))) float v8f;

// ---------------------------------------------------------------------------
// Epilogue + store of one 16x16 f32 accumulator tile.
//   EPI: 0=raw  1=bias+prelu(alpha)  2=bias+relu  3=bias+prelu(alpha)+residual
// C/D layout (ISA 7.12.2): lane L, VGPR v -> C[v + 8*(L>>4), L&15]
// ---------------------------------------------------------------------------
template <int EPI>
__device__ __forceinline__ void epilogue_store(const v8f& acc, float* __restrict__ C,
                                               int ldc, int row0, int cN, int M, int N,
                                               int lane, const float* __restrict__ bias,
                                               float al, const float* __restrict__ resid,
                                               long residOff) {
  if (cN >= N) return;
  float bv = 0.0f;
  if constexpr (EPI != 0) bv = bias[cN];
#pragma unroll
  for (int v = 0; v < 8; ++v) {
    const int m = row0 + v + 8 * (lane >> 4);
    if (m < M) {
      float x = acc[v];
      if constexpr (EPI != 0) x += bv;
      if constexpr (EPI == 2) x = fmaxf(x, 0.0f);
      if constexpr (EPI == 1 || EPI == 3) x = (x >= 0.0f) ? x : al * x;
      if constexpr (EPI == 3) x += resid[residOff + (long)m * ldc + cN];
      C[(long)m * ldc + cN] = x;
    }
  }
}

// ---------------------------------------------------------------------------
// fp32 WMMA GEMM, 32x32 tile per wave32 (2x2 subtiles, 4 accumulators).
//   TRANSB=0: C[m,n] = sum_k A[m,k]*B[k,n]   (A,B row-major)
//   TRANSB=1: C[m,n] = sum_k A[m,k]*B[n,k]   (A,B row-major, K contiguous)
// Edge tiles: fragment indices are CLAMPED (no zeroing, no exec divergence) —
// out-of-range rows/cols of A/B only pollute C rows/cols we never store.
// Inner loop: unconditional b64/b32 loads + 4 independent v_wmma per K-step.
// A-frag (ISA 7.12.2): lane L holds A[L&15, k0 + 2*(L>>4) + v], v=0,1.
// ---------------------------------------------------------------------------
template <bool TRANSB, int EPI>
__global__ void gemm_wmma_f32(const float* __restrict__ A, int lda, long strideA,
                              const float* __restrict__ B, int ldb, long strideB,
                              float* __restrict__ C, int ldc, long strideC,
                              int M, int N, int K,
                              const float* __restrict__ bias,
                              const float* __restrict__ alpha,
                              const float* __restrict__ resid, long strideR) {
  const int lane = threadIdx.x;          // 0..31, wave32
  const int row0 = blockIdx.x * 32;
  const int col0 = blockIdx.y * 32;
  const int bz   = blockIdx.z;
  A += (long)bz * strideA;
  B += (long)bz * strideB;
  C += (long)bz * strideC;

  const int mfrag = lane & 15;
  const int khalf = (lane >> 4) * 2;     // 0 or 2

  const int rA0 = row0 + mfrag, rA1 = row0 + 16 + mfrag;
  const int cB0 = col0 + mfrag, cB1 = col0 + 16 + mfrag;
  const int r0 = (rA0 < M) ? rA0 : (M - 1);     // clamped fragment indices
  const int r1 = (rA1 < M) ? rA1 : (M - 1);
  const int c0 = (cB0 < N) ? cB0 : (N - 1);
  const int c1 = (cB1 < N) ? cB1 : (N - 1);

  const float* A0 = A + (long)r0 * lda + khalf;
  const float* A1 = A + (long)r1 * lda + khalf;
  const float* B0;
  const float* B1;
  if (TRANSB) {
    B0 = B + (long)c0 * ldb + khalf;
    B1 = B + (long)c1 * ldb + khalf;
  } else {
    B0 = B + (long)khalf * ldb + c0;
    B1 = B + (long)khalf * ldb + c1;
  }

  v8f acc00 = {}, acc01 = {}, acc10 = {}, acc11 = {};
  for (int k0 = 0; k0 < K; k0 += 4) {
    const v2f a0 = *(const v2f*)A0;
    const v2f a1 = *(const v2f*)A1;
    v2f b0, b1;
    if (TRANSB) {
      b0 = *(const v2f*)B0;
      b1 = *(const v2f*)B1;
      B0 += 4; B1 += 4;
    } else {
      b0[0] = B0[0]; b0[1] = B0[ldb];
      b1[0] = B1[0]; b1[1] = B1[ldb];
      B0 += 4 * (long)ldb; B1 += 4 * (long)ldb;
    }
    A0 += 4; A1 += 4;
    acc00 = __builtin_amdgcn_wmma_f32_16x16x4_f32(false, a0, false, b0, (short)0, acc00, false, false);
    acc01 = __builtin_amdgcn_wmma_f32_16x16x4_f32(false, a0, false, b1, (short)0, acc01, false, false);
    acc10 = __builtin_amdgcn_wmma_f32_16x16x4_f32(false, a1, false, b0, (short)0, acc10, false, false);
    acc11 = __builtin_amdgcn_wmma_f32_16x16x4_f32(false, a1, false, b1, (short)0, acc11, false, false);
  }

  float al = 0.0f;
  if constexpr (EPI == 1 || EPI == 3) al = alpha[0];
  const long rOff = (long)bz * strideR;
  epilogue_store<EPI>(acc00, C, ldc, row0,      cB0, M, N, lane, bias, al, resid, rOff);
  epilogue_store<EPI>(acc01, C, ldc, row0,      cB1, M, N, lane, bias, al, resid, rOff);
  epilogue_store<EPI>(acc10, C, ldc, row0 + 16, cB0, M, N, lane, bias, al, resid, rOff);
  epilogue_store<EPI>(acc11, C, ldc, row0 + 16, cB1, M, N, lane, bias, al, resid, rOff);
}

// ---------------------------------------------------------------------------
// Row softmax (contiguous rows of length `cols`), one 256-thread block / row.
// ---------------------------------------------------------------------------
__global__ void softmax_rows(float* __restrict__ data, int cols) {
  float* p = data + (long)blockIdx.x * cols;
  __shared__ float red[256];
  const int t = threadIdx.x;

  float m = -INFINITY;
  for (int j = t; j < cols; j += 256) m = fmaxf(m, p[j]);
  red[t] = m; __syncthreads();
  for (int s = 128; s > 0; s >>= 1) { if (t < s) red[t] = fmaxf(red[t], red[t + s]); __syncthreads(); }
  m = red[0]; __syncthreads();

  float sum = 0.0f;
  for (int j = t; j < cols; j += 256) { float e = __expf(p[j] - m); p[j] = e; sum += e; }
  red[t] = sum; __syncthreads();
  for (int s = 128; s > 0; s >>= 1) { if (t < s) red[t] += red[t + s]; __syncthreads(); }
  const float inv = 1.0f / red[0];
  for (int j = t; j < cols; j += 256) p[j] *= inv;
}

// Bm[i,j] = (i==j) ? 1 : sqrt(max(PPt[i,i]+PPt[j,j]-2*PPt[i,j], 0))
__global__ void bm_from_ppt(const float* __restrict__ ppt, float* __restrict__ bm, int S) {
  const long base = (long)blockIdx.z * S * S;
  const int i = blockIdx.y;
  const int j = blockIdx.x * blockDim.x + threadIdx.x;
  if (j >= S) return;
  float v;
  if (i == j) v = 1.0f;
  else {
    float d2 = ppt[base + (long)i * S + i] + ppt[base + (long)j * S + j]
             - 2.0f * ppt[base + (long)i * S + j];
    v = sqrtf(fmaxf(d2, 0.0f));
  }
  bm[base + (long)i * S + j] = v;
}

// dh[row] = rsqrt(sum_j Bm[row, j])   (rows = B*S, contiguous rows)
__global__ void rowsum_rsqrt(const float* __restrict__ bm, float* __restrict__ dh,
                             int rows, int S) {
  const int row = blockIdx.x * blockDim.x + threadIdx.x;
  if (row >= rows) return;
  const float* p = bm + (long)row * S;
  float s = 0.0f;
  for (int j = 0; j < S; ++j) s += p[j];
  dh[row] = rsqrtf(s);
}

// M[i,j] = dh[i] * Bm[i,j] * dh[j]
__global__ void scale_m(const float* __restrict__ bm, const float* __restrict__ dh,
                        float* __restrict__ mm, int S) {
  const int bz = blockIdx.z;
  const long base = (long)bz * S * S;
  const int i = blockIdx.y;
  const int j = blockIdx.x * blockDim.x + threadIdx.x;
  if (j >= S) return;
  mm[base + (long)i * S + j] =
      dh[(long)bz * S + i] * bm[base + (long)i * S + j] * dh[(long)bz * S + j];
}

// Fused h / logits: avoids materializing the (B,N,S,D) tensor.
// logits[b,n,s] = prelu( sum_d prelu(h1[b,n,d]+h2[b,s,d]+b4a[d], a4a)*W4b[d] + b4b, a4b )
__global__ void logits_fused(const float* __restrict__ h1, const float* __restrict__ h2,
                             const float* __restrict__ b4a, const float* __restrict__ W4b,
                             const float* __restrict__ b4b, const float* __restrict__ a4a_p,
                             const float* __restrict__ a4b_p, float* __restrict__ logits,
                             int Nn, int S, int D) {
  const int bs = blockIdx.x;         // b*S + s
  const int b = bs / S, s = bs % S;
  __shared__ float sh2[128], sw[128], sb[128];
  const int t = threadIdx.x;
  for (int d = t; d < D; d += blockDim.x) {
    sh2[d] = h2[((long)b * S + s) * D + d];
    sw[d]  = W4b[d];
    sb[d]  = b4a[d];
  }
  __syncthreads();
  const float a4a = a4a_p[0], a4b = a4b_p[0], bb = b4b[0];
  for (int n = t; n < Nn; n += blockDim.x) {
    const float* h1p = h1 + ((long)b * Nn + n) * D;
    float acc = 0.0f;
    for (int d = 0; d < D; ++d) {
      float v = h1p[d] + sh2[d] + sb[d];
      v = (v >= 0.0f) ? v : a4a * v;
      acc = fmaf(v, sw[d], acc);
    }
    acc += bb;
    acc = (acc >= 0.0f) ? acc : a4b * acc;
    logits[((long)b * Nn + n) * S + s] = acc;
  }
}

// Softmax over N (axis=1) of (B,N,S): one thread per (b,s) column.
// Adjacent threads = adjacent s -> every pass is coalesced.
__global__ void softmax_cols(float* __restrict__ data, int B, int Nn, int S) {
  const long idx = (long)blockIdx.x * blockDim.x + threadIdx.x;
  if (idx >= (long)B * S) return;
  const int b = (int)(idx / S), s = (int)(idx % S);
  float* p = data + (long)b * Nn * S + s;
  float m = -INFINITY;
  for (int n = 0; n < Nn; ++n) m = fmaxf(m, p[(long)n * S]);
  float sum = 0.0f;
  for (int n = 0; n < Nn; ++n) { float e = __expf(p[(long)n * S] - m); p[(long)n * S] = e; sum += e; }
  const float inv = 1.0f / sum;
  for (int n = 0; n < Nn; ++n) p[(long)n * S] *= inv;
}

// ---------------------------------------------------------------------------
extern "C" void kernel_launch(void* const* d_in, const int* in_sizes, int n_in,
                              void* d_out, int out_size, void* d_ws, size_t ws_size,
                              hipStream_t stream) {
  (void)in_sizes; (void)n_in; (void)out_size; (void)ws_size;

  constexpr int Bn = 4, Nn = 400, S = 512, D = 100;
  constexpr long SS = (long)S * S;          // 262144
  constexpr long ND = (long)Nn * D;         // 40000
  constexpr long SD = (long)S * D;          // 51200

  const float* x    = (const float*)d_in[0];
  const float* G    = (const float*)d_in[1];
  const float* Am   = (const float*)d_in[2];
  const float* W_gf = (const float*)d_in[3];
  const float* b_gf = (const float*)d_in[4];
  const float* a_gf = (const float*)d_in[5];
  const float* Wg   = (const float*)d_in[6];
  const float* bg   = (const float*)d_in[7];
  const float* W4a  = (const float*)d_in[8];
  const float* b4a  = (const float*)d_in[9];
  const float* a4a  = (const float*)d_in[10];
  const float* W4b  = (const float*)d_in[11];
  const float* b4b  = (const float*)d_in[12];
  const float* a4b  = (const float*)d_in[13];
  const float* W5   = (const float*)d_in[14];
  const float* b5   = (const float*)d_in[15];
  const float* a5   = (const float*)d_in[16];
  float* out = (float*)d_out;

  float* ws    = (float*)d_ws;
  float* bufP  = ws;                        // B*S*S : P, later logits/div
  float* buf1  = ws + 4 * SS;               // B*S*S : Bm, later h1|h2
  float* buf2  = ws + 8 * SS;               // B*S*S : PPt -> M -> adjP
  float* buf3  = ws + 12 * SS;              // B*S*S : adj, later t=div@gx
  float* G1    = ws + 16 * SS;              // S*D
  float* Axb   = G1 + SD;                   // B*S*D
  float* gx    = Axb + (long)Bn * SD;       // B*S*D
  float* dh    = gx + (long)Bn * SD;        // B*S
  float* h1    = buf1;                      // B*N*D
  float* h2    = buf1 + (long)Bn * ND;      // B*S*D

  const dim3 blk(32);

  // 1) G1 = prelu(G @ W_gf^T + b_gf, a_gf)          [512x100, K=400]
  gemm_wmma_f32<true, 1><<<dim3(16, 4, 1), blk, 0, stream>>>(
      G, 400, 0, W_gf, 400, 0, G1, D, 0, S, D, Nn, b_gf, a_gf, nullptr, 0);

  // 2) Ax = A @ x   per batch                        [512x100, K=400]
  gemm_wmma_f32<false, 0><<<dim3(16, 4, Bn), blk, 0, stream>>>(
      Am, 400, 0, x, D, ND, Axb, D, SD, S, D, Nn, nullptr, nullptr, nullptr, 0);

  // 3) G_AX = G1 @ Ax^T -> P buffer                  [512x512, K=100]
  gemm_wmma_f32<true, 0><<<dim3(16, 16, Bn), blk, 0, stream>>>(
      G1, D, 0, Axb, D, SD, bufP, S, SS, S, S, D, nullptr, nullptr, nullptr, 0);

  // 4) P = softmax over last axis
  softmax_rows<<<dim3(Bn * S), dim3(256), 0, stream>>>(bufP, S);

  // 5) PPt = P @ P^T                                 [512x512, K=512]
  gemm_wmma_f32<true, 0><<<dim3(16, 16, Bn), blk, 0, stream>>>(
      bufP, S, SS, bufP, S, SS, buf2, S, SS, S, S, S, nullptr, nullptr, nullptr, 0);

  // 6) Bm = dist + I  (from PPt; diag(PPt)=sq)
  bm_from_ppt<<<dim3(S / 256, S, Bn), dim3(256), 0, stream>>>(buf2, buf1, S);

  // 7) dh = rsqrt(rowsum(Bm))
  rowsum_rsqrt<<<dim3((Bn * S + 255) / 256), dim3(256), 0, stream>>>(buf1, dh, Bn * S, S);

  // 8) M = dh (.) Bm (.) dh
  scale_m<<<dim3(S / 256, S, Bn), dim3(256), 0, stream>>>(buf1, dh, buf2, S);

  // 9) adj = M @ Bm                                  [512x512, K=512]
  gemm_wmma_f32<false, 0><<<dim3(16, 16, Bn), blk, 0, stream>>>(
      buf2, S, SS, buf1, S, SS, buf3, S, SS, S, S, S, nullptr, nullptr, nullptr, 0);

  // 10) adjP = adj @ P                               [512x512, K=512]
  gemm_wmma_f32<false, 0><<<dim3(16, 16, Bn), blk, 0, stream>>>(
      buf3, S, SS, bufP, S, SS, buf2, S, SS, S, S, S, nullptr, nullptr, nullptr, 0);

  // 11) gx = relu(adjP @ Wg^T + bg)                  [512x100, K=512]
  gemm_wmma_f32<true, 2><<<dim3(16, 4, Bn), blk, 0, stream>>>(
      buf2, S, SS, Wg, S, 0, gx, D, SD, S, D, S, bg, nullptr, nullptr, 0);

  // 12) h1 = x @ W4a[:, :D]^T                        [400x100, K=100]
  gemm_wmma_f32<true, 0><<<dim3(13, 4, Bn), blk, 0, stream>>>(
      x, D, ND, W4a, 2 * D, 0, h1, D, ND, Nn, D, D, nullptr, nullptr, nullptr, 0);

  // 13) h2 = gx @ W4a[:, D:]^T                       [512x100, K=100]
  gemm_wmma_f32<true, 0><<<dim3(16, 4, Bn), blk, 0, stream>>>(
      gx, D, SD, W4a + D, 2 * D, 0, h2, D, SD, S, D, D, nullptr, nullptr, nullptr, 0);

  // 14) logits (fused prelu-dot-prelu; never materialize (B,N,S,D))
  logits_fused<<<dim3(Bn * S), dim3(256), 0, stream>>>(
      h1, h2, b4a, W4b, b4b, a4a, a4b, bufP, Nn, S, D);

  // 15) div = softmax over N
  softmax_cols<<<dim3((Bn * S + 255) / 256), dim3(256), 0, stream>>>(bufP, Bn, Nn, S);

  // 16) t = div @ gx                                 [400x100, K=512]
  gemm_wmma_f32<false, 0><<<dim3(13, 4, Bn), blk, 0, stream>>>(
      bufP, S, (long)Nn * S, gx, D, SD, buf3, D, ND, Nn, D, S,
      nullptr, nullptr, nullptr, 0);

  // 17) out = prelu(t @ W5^T + b5, a5) + x           [400x100, K=100]
  gemm_wmma_f32<true, 3><<<dim3(13, 4, Bn), blk, 0, stream>>>(
      buf3, D, ND, W5, D, 0, out, D, ND, Nn, D, D, b5, a5, x, ND);
}